// GCNEncoder_87316685127958
// MI455X (gfx1250) — compile-verified
//
#include <hip/hip_runtime.h>

typedef __attribute__((ext_vector_type(2))) float v2f;
typedef __attribute__((ext_vector_type(8))) float v8f;

#define D 128
#define WPAD 132  // LDS row stride; WPAD % 64 == 4 -> conflict-free b64 fragment loads

// ---------------------------------------------------------------------------
// H[N x 128] = X[N x 128] @ W^T   (W is [128 x 128] row-major, h[i][o] = sum_k x[i][k] W[o][k])
// Block: 256 threads = 8 waves. Wave w computes rows rowBase..rowBase+15, all 8 col tiles.
// Uses V_WMMA_F32_16X16X4_F32, K-loop of 32 steps.
// ---------------------------------------------------------------------------
__global__ __launch_bounds__(256) void gemm_xWt(const float* __restrict__ X,
                                                const float* __restrict__ W,
                                                float* __restrict__ H, int n) {
    extern __shared__ float sW[];  // 128 * WPAD floats = 67,584 B
    const int tid = threadIdx.x;

    // Cooperatively stage W (128x128 f32) into padded LDS, float4 coalesced loads.
    for (int i = tid; i < 128 * 32; i += 256) {
        int r  = i >> 5;        // row 0..127
        int c4 = i & 31;        // float4 column 0..31
        float4 w = ((const float4*)W)[r * 32 + c4];
        float* p = &sW[r * WPAD + c4 * 4];
        p[0] = w.x; p[1] = w.y; p[2] = w.z; p[3] = w.w;
    }
    __syncthreads();

    const int wave = tid >> 5;
    const int lane = tid & 31;
    const int lr   = lane & 15;   // row-in-tile (A) / col-in-tile (B,D)
    const int lh   = lane >> 4;   // half-wave select
    const int rowBase = blockIdx.x * 128 + wave * 16;

    // A fragment rows: lanes 0-15 hold rows 0..15 (K,K+1), lanes 16-31 same rows (K+2,K+3).
    int arow = rowBase + lr;
    if (arow > n - 1) arow = n - 1;  // clamp OOB rows; stores are predicated below

    // Preload all 32 A fragments (K=128 / 4) into registers: 32 x global_load_b64.
    v2f a[32];
    const float* aptr = X + (size_t)arow * D + 2 * lh;
    #pragma unroll
    for (int kk = 0; kk < 32; ++kk)
        a[kk] = *(const v2f*)(aptr + 4 * kk);

    v8f acc[8] = {};

    for (int kk = 0; kk < 32; ++kk) {
        // B fragment: B[k][nn] = W[nn][k]; lane n<16 -> (W[16c+n][4kk], W[16c+n][4kk+1]),
        // lanes 16-31 -> K+2,K+3. Conflict-free thanks to WPAD%64==4.
        const float* bbase = &sW[lr * WPAD + 4 * kk + 2 * lh];
        #pragma unroll
        for (int c = 0; c < 8; ++c) {
            v2f b = *(const v2f*)(bbase + c * 16 * WPAD);
            acc[c] = __builtin_amdgcn_wmma_f32_16x16x4_f32(
                false, a[kk], false, b, (short)0, acc[c], false, false);
        }
    }

    // D layout: VGPR r holds M = r (lanes 0-15) / r+8 (lanes 16-31); N = lane&15.
    #pragma unroll
    for (int c = 0; c < 8; ++c) {
        #pragma unroll
        for (int r = 0; r < 8; ++r) {
            int row = rowBase + r + 8 * lh;
            if (row < n)
                H[(size_t)row * D + c * 16 + lr] = acc[c][r];
        }
    }
}

// ---------------------------------------------------------------------------
// Degree + normalization
// ---------------------------------------------------------------------------
__global__ void degree_k(const int* __restrict__ dst, int* __restrict__ deg, int e) {
    int i = blockIdx.x * blockDim.x + threadIdx.x;
    if (i < e) atomicAdd(&deg[dst[i]], 1);
}

__global__ void dinv_k(const int* __restrict__ deg, float* __restrict__ dinv, int n) {
    int i = blockIdx.x * blockDim.x + threadIdx.x;
    if (i < n) dinv[i] = rsqrtf((float)deg[i] + 1.0f);
}

// ---------------------------------------------------------------------------
// Edge aggregation: one wave32 per edge; lane handles 4 channels (float4).
// agg[dst] += dinv[src]*dinv[dst] * h[src].  h and agg are L2-resident (51.2 MB each).
// ---------------------------------------------------------------------------
__global__ __launch_bounds__(256) void aggregate_k(const float* __restrict__ H,
                                                   const int* __restrict__ src,
                                                   const int* __restrict__ dst,
                                                   const float* __restrict__ dinv,
                                                   float* __restrict__ Agg, int e) {
    int eidx = blockIdx.x * 8 + (threadIdx.x >> 5);
    if (eidx >= e) return;
    int lane = threadIdx.x & 31;
    int s = src[eidx];
    int d = dst[eidx];
    float norm = dinv[s] * dinv[d];
    float4 v = ((const float4*)(H + (size_t)s * D))[lane];
    float* out = Agg + (size_t)d * D + lane * 4;
    unsafeAtomicAdd(out + 0, v.x * norm);
    unsafeAtomicAdd(out + 1, v.y * norm);
    unsafeAtomicAdd(out + 2, v.z * norm);
    unsafeAtomicAdd(out + 3, v.w * norm);
}

// ---------------------------------------------------------------------------
// Self-loop + bias (in place) and optional per-channel BN statistics.
// block = 128 threads (thread = channel), grid-stride over rows, one atomic/thread.
// ---------------------------------------------------------------------------
__global__ __launch_bounds__(128) void selfloop_bias_stats_k(const float* __restrict__ H,
                                                             const float* __restrict__ dinv,
                                                             const float* __restrict__ bias,
                                                             float* __restrict__ Agg,
                                                             float* __restrict__ sum,
                                                             float* __restrict__ sumsq,
                                                             int n, int doStats) {
    int ch = threadIdx.x;
    float b = bias[ch];
    float acc = 0.f, acc2 = 0.f;
    for (int row = blockIdx.x; row < n; row += gridDim.x) {
        float di = dinv[row];
        size_t off = (size_t)row * D + ch;
        float v = Agg[off] + di * di * H[off] + b;
        Agg[off] = v;
        acc  += v;
        acc2 += v * v;
    }
    if (doStats) {
        unsafeAtomicAdd(&sum[ch], acc);
        unsafeAtomicAdd(&sumsq[ch], acc2);
    }
}

// ---------------------------------------------------------------------------
// Per-channel affine for BN: alpha = g*rsqrt(var+eps), bet = beta - mu*alpha
// ---------------------------------------------------------------------------
__global__ void bn_affine_k(const float* __restrict__ sum, const float* __restrict__ sumsq,
                            const float* __restrict__ g, const float* __restrict__ beta,
                            float* __restrict__ alpha, float* __restrict__ bet, float invN) {
    int ch = threadIdx.x;
    float mu  = sum[ch] * invN;
    float var = sumsq[ch] * invN - mu * mu;  // biased variance, matches jnp.var
    float a = g[ch] * rsqrtf(var + 1e-5f);
    alpha[ch] = a;
    bet[ch]   = beta[ch] - mu * a;
}

// ---------------------------------------------------------------------------
// Fused BN scale/shift + ReLU, float4 vectorized.
// ---------------------------------------------------------------------------
__global__ __launch_bounds__(256) void bn_relu_k(const float* __restrict__ in,
                                                 const float* __restrict__ alpha,
                                                 const float* __restrict__ bet,
                                                 float* __restrict__ out, size_t n4) {
    size_t i = (size_t)blockIdx.x * blockDim.x + threadIdx.x;
    if (i >= n4) return;
    int ch = ((int)(i & 31)) * 4;  // 32 float4 per row of 128 channels
    float4 v = ((const float4*)in)[i];
    v.x = fmaxf(v.x * alpha[ch + 0] + bet[ch + 0], 0.f);
    v.y = fmaxf(v.y * alpha[ch + 1] + bet[ch + 1], 0.f);
    v.z = fmaxf(v.z * alpha[ch + 2] + bet[ch + 2], 0.f);
    v.w = fmaxf(v.w * alpha[ch + 3] + bet[ch + 3], 0.f);
    ((float4*)out)[i] = v;
}

// ---------------------------------------------------------------------------
extern "C" void kernel_launch(void* const* d_in, const int* in_sizes, int n_in,
                              void* d_out, int out_size, void* d_ws, size_t ws_size,
                              hipStream_t stream) {
    const float* x   = (const float*)d_in[0];
    const int*   ei  = (const int*)d_in[1];
    const float* W1  = (const float*)d_in[2];
    const float* b1  = (const float*)d_in[3];
    const float* g1  = (const float*)d_in[4];
    const float* be1 = (const float*)d_in[5];
    const float* W2  = (const float*)d_in[6];
    const float* b2  = (const float*)d_in[7];
    const float* g2  = (const float*)d_in[8];
    const float* be2 = (const float*)d_in[9];
    const float* W3  = (const float*)d_in[10];
    const float* b3  = (const float*)d_in[11];

    const int N = in_sizes[0] / D;
    const int E = in_sizes[1] / 2;
    const int* src = ei;
    const int* dst = ei + E;

    const size_t nd = (size_t)N * D;
    float* bufA  = (float*)d_ws;        // 51.2 MB
    float* bufB  = bufA + nd;           // 51.2 MB
    float* dinv  = bufB + nd;           // N floats
    int*   deg   = (int*)(dinv + N);    // N ints
    float* sum   = (float*)(deg + N);   // 128
    float* sumsq = sum + D;             // 128
    float* alpha = sumsq + D;           // 128
    float* bet   = alpha + D;           // 128
    float* outp  = (float*)d_out;

    const int smem = 128 * WPAD * 4;    // 67,584 B dynamic LDS (well under 320 KB/WGP)
    const dim3 blk256(256), blk128(128);
    const int gGemm = (N + 127) / 128;
    const int gEdge = (E + 7) / 8;
    const int gElem = (int)((nd / 4 + 255) / 256);

    // --- graph normalization (same for all 3 layers) ---
    hipMemsetAsync(deg, 0, (size_t)N * sizeof(int), stream);
    degree_k<<<(E + 255) / 256, blk256, 0, stream>>>(dst, deg, E);
    dinv_k<<<(N + 255) / 256, blk256, 0, stream>>>(deg, dinv, N);

    // --- layer 1: x -> bufA(H) -> bufB(agg) -> bufA(relu(bn)) ---
    gemm_xWt<<<gGemm, blk256, smem, stream>>>(x, W1, bufA, N);
    hipMemsetAsync(bufB, 0, nd * sizeof(float), stream);
    aggregate_k<<<gEdge, blk256, 0, stream>>>(bufA, src, dst, dinv, bufB, E);
    hipMemsetAsync(sum, 0, 2 * D * sizeof(float), stream);
    selfloop_bias_stats_k<<<1024, blk128, 0, stream>>>(bufA, dinv, b1, bufB, sum, sumsq, N, 1);
    bn_affine_k<<<1, blk128, 0, stream>>>(sum, sumsq, g1, be1, alpha, bet, 1.0f / (float)N);
    bn_relu_k<<<gElem, blk256, 0, stream>>>(bufB, alpha, bet, bufA, nd / 4);

    // --- layer 2: bufA -> bufB(H) -> bufA(agg) -> bufB(relu(bn)) ---
    gemm_xWt<<<gGemm, blk256, smem, stream>>>(bufA, W2, bufB, N);
    hipMemsetAsync(bufA, 0, nd * sizeof(float), stream);
    aggregate_k<<<gEdge, blk256, 0, stream>>>(bufB, src, dst, dinv, bufA, E);
    hipMemsetAsync(sum, 0, 2 * D * sizeof(float), stream);
    selfloop_bias_stats_k<<<1024, blk128, 0, stream>>>(bufB, dinv, b2, bufA, sum, sumsq, N, 1);
    bn_affine_k<<<1, blk128, 0, stream>>>(sum, sumsq, g2, be2, alpha, bet, 1.0f / (float)N);
    bn_relu_k<<<gElem, blk256, 0, stream>>>(bufA, alpha, bet, bufB, nd / 4);

    // --- layer 3: bufB -> bufA(H) -> d_out(agg + b3) ---
    gemm_xWt<<<gGemm, blk256, smem, stream>>>(bufB, W3, bufA, N);
    hipMemsetAsync(outp, 0, nd * sizeof(float), stream);
    aggregate_k<<<gEdge, blk256, 0, stream>>>(bufA, src, dst, dinv, outp, E);
    selfloop_bias_stats_k<<<1024, blk128, 0, stream>>>(bufA, dinv, b3, outp, nullptr, nullptr, N, 0);
}